// LowRankSTLayer_75960791597925
// MI455X (gfx1250) — compile-verified
//
#include <hip/hip_runtime.h>
#include <cstdint>

typedef __attribute__((ext_vector_type(2))) float v2f;
typedef __attribute__((ext_vector_type(8))) float v8f;
typedef __attribute__((ext_vector_type(4))) unsigned int u32x4;
typedef __attribute__((ext_vector_type(8))) int i32x8;
typedef __attribute__((ext_vector_type(4))) int i32x4;

#define B_   2
#define C_   64
#define D_   8
#define H_   128
#define W_   128
#define HW_  (H_*W_)      // 16384
#define DHW_ (D_*HW_)     // 131072
#define NVOX (B_*DHW_)    // 262144

#define WPITCH 72         // weight LDS pitch: banks (8k+n)%64, halves disjoint
#define XPITCH 144        // x-tile LDS pitch: high half shifted 32 banks
#define UVPITCH 68        // UV LDS pitch: read banks {4m} vs {4m+2} disjoint

// ---------------------------------------------------------------------------
// Pass 1: h[b,o,v] = sum_i head_w[o,i] * x[b,i,v]   (1x1x1 conv as GEMM)
// x tile staged LDS-direct with GLOBAL_LOAD_ASYNC_TO_LDS_B128 (ASYNCcnt),
// each HBM byte read exactly once. Each wave: 1 M-tile x 4 N-tiles,
// 64 x v_wmma_f32_16x16x4_f32.
// ---------------------------------------------------------------------------
__global__ __launch_bounds__(256) void head_gemm_kernel(
    const float* __restrict__ x, const float* __restrict__ head_w,
    float* __restrict__ h)
{
  __shared__ float wT[C_ * WPITCH];   // head_w transposed [k][n]
  __shared__ float xt[C_ * XPITCH];   // x tile [k][v]

  const int t = threadIdx.x;

  const long vox0 = (long)blockIdx.x * 128;
  const int  b    = (int)(vox0 >> 17);        // 131072 voxels per batch
  const long vloc = vox0 & (DHW_ - 1);
  const float* xb = x + (long)b * C_ * DHW_ + vloc;

  // async b128 staging: global -> LDS directly, no VGPR round trip
  #pragma unroll
  for (int i = t * 4; i < C_ * 128; i += 1024) {
    const int c = i >> 7, v = i & 127;
    const unsigned lds = (unsigned)(uintptr_t)&xt[c * XPITCH + v];
    const unsigned long long ga =
        (unsigned long long)(uintptr_t)(xb + (long)c * DHW_ + v);
    asm volatile("global_load_async_to_lds_b128 %0, %1, off"
                 :: "v"(lds), "v"(ga) : "memory");
  }
  // weight staging overlaps the async DMA
  for (int i = t; i < C_ * C_; i += 256) {
    const int n = i >> 6, k = i & 63;
    wT[k * WPITCH + n] = head_w[i];
  }
  asm volatile("s_wait_asynccnt 0x0" ::: "memory");
  __syncthreads();

  const int wave = t >> 5, lane = t & 31;
  const int half = lane >> 4;          // 0: K=k0,k0+1  1: K=k0+2,k0+3
  const int mrow = lane & 15;
  const int av   = wave * 16 + mrow;   // A row voxel within the 128-tile

  const v8f zero = {0.f, 0.f, 0.f, 0.f, 0.f, 0.f, 0.f, 0.f};
  v8f acc[4];
  #pragma unroll
  for (int nt = 0; nt < 4; ++nt) acc[nt] = zero;

  #pragma unroll
  for (int k0 = 0; k0 < C_; k0 += 4) {
    const int ka = k0 + 2 * half;
    v2f a;
    a.x = xt[ka * XPITCH + av];
    a.y = xt[(ka + 1) * XPITCH + av];
    #pragma unroll
    for (int nt = 0; nt < 4; ++nt) {
      v2f bm;
      bm.x = wT[ka * WPITCH + nt * 16 + mrow];        // B[k,n] = head_w[n,k]
      bm.y = wT[(ka + 1) * WPITCH + nt * 16 + mrow];
      acc[nt] = __builtin_amdgcn_wmma_f32_16x16x4_f32(
          false, a, false, bm, (short)0, acc[nt], false, false);
    }
  }
  // D: VGPR j -> (M = j + 8*half, N = lane&15); M tile = voxels wave*16..+15
  #pragma unroll
  for (int nt = 0; nt < 4; ++nt) {
    const int ncol = nt * 16 + mrow;
    float* hp = h + ((long)b * C_ + ncol) * DHW_ + vloc + wave * 16 + 8 * half;
    #pragma unroll
    for (int j = 0; j < 8; ++j) hp[j] = acc[nt][j];
  }
}

// ---------------------------------------------------------------------------
// Pass 2: per 8x16 spatial tile (fixed b,d):
//   Vp[r] = h . V_w, Up = 3x3x3 depthwise (halo window), Un = l2norm(rank),
//   UV = <Un,Vp>, out = tail_w @ UV (WMMA f32 16x16x4).
// Interior tiles: halo window is an 18x10x3 strided tile -> Tensor Data Mover
// (double-buffered, TENSORcnt). Boundary tiles: clamped gather via
// GLOBAL_LOAD_ASYNC_TO_LDS_B32 (ASYNCcnt). h (67 MB) is L2-resident.
// ---------------------------------------------------------------------------
__global__ __launch_bounds__(256) void fused_lowrank_tail_kernel(
    const float* __restrict__ h, const float* __restrict__ tail_w,
    const float* __restrict__ U_w, const float* __restrict__ V_w,
    float* __restrict__ out)
{
  __shared__ float tT[C_ * WPITCH];        // tail_w transposed [k][n]
  __shared__ float uw[96];                 // U_w 27x3 (padded)
  __shared__ float vw[C_ * 3];             // V_w 64x3
  __shared__ float win[2][2][3 * 10 * 18]; // [parity][channel] halo windows
  __shared__ float vpp[2][3][128];         // Vp partial sums
  __shared__ float uvt[128 * UVPITCH];     // UV tile

  const int t = threadIdx.x;
  for (int i = t; i < C_ * C_; i += 256) {
    const int n = i >> 6, k = i & 63;
    tT[k * WPITCH + n] = tail_w[i];
  }
  if (t < 81)     uw[t] = U_w[t];
  if (t < C_ * 3) vw[t] = V_w[t];

  const int bid = blockIdx.x;
  const int xt_ = bid & 7;              // W/16 = 8 tiles
  const int yt  = (bid >> 3) & 15;      // H/8  = 16 tiles
  const int d0  = (bid >> 7) & 7;
  const int b   = bid >> 10;
  const int x0 = xt_ * 16, y0 = yt * 8;
  const bool interior = (d0 >= 1) & (d0 <= 6) & (yt >= 1) & (yt <= 14) &
                        (xt_ >= 1) & (xt_ <= 6);

  const int vox   = t & 127;            // voxel within tile
  const int chalf = t >> 7;             // channel-split half
  const int ly = vox >> 4, lx = vox & 15;

  const long center = (long)d0 * HW_ + (long)(y0 + ly) * W_ + (x0 + lx);
  const float* hb = h + (long)b * C_ * DHW_;

  __syncthreads();

  // ---- Phase A: Vp partials over 32 channels per thread-half ----
  {
    float a0 = 0.f, a1 = 0.f, a2 = 0.f;
    const int cbase = chalf * 32;
    const float* hc = hb + (long)cbase * DHW_ + center;
    #pragma unroll 4
    for (int c = 0; c < 32; ++c) {
      const float hv = hc[(long)c * DHW_];
      const float* v = &vw[(cbase + c) * 3];
      a0 = fmaf(hv, v[0], a0);
      a1 = fmaf(hv, v[1], a1);
      a2 = fmaf(hv, v[2], a2);
    }
    vpp[chalf][0][vox] = a0;
    vpp[chalf][1][vox] = a1;
    vpp[chalf][2][vox] = a2;
  }
  __syncthreads();
  const float vp0 = vpp[0][0][vox] + vpp[1][0][vox];
  const float vp1 = vpp[0][1][vox] + vpp[1][1][vox];
  const float vp2 = vpp[0][2][vox] + vpp[1][2][vox];

  // depthwise compute for one channel window -> writes uvt[vox][c]
  auto dw_compute = [&](const float* w, int c) {
    float up0 = 0.f, up1 = 0.f, up2 = 0.f;
    #pragma unroll
    for (int kz = 0; kz < 3; ++kz)
      #pragma unroll
      for (int ky = 0; ky < 3; ++ky)
        #pragma unroll
        for (int kx = 0; kx < 3; ++kx) {
          const float hv = w[kz * 180 + (ly + ky) * 18 + (lx + kx)];
          const float* u = &uw[((kz * 3 + ky) * 3 + kx) * 3];
          up0 = fmaf(hv, u[0], up0);
          up1 = fmaf(hv, u[1], up1);
          up2 = fmaf(hv, u[2], up2);
        }
    const float nrm = sqrtf(up0 * up0 + up1 * up1 + up2 * up2);
    const float inv = 1.0f / (1e-6f + nrm);
    uvt[vox * UVPITCH + c] = (up0 * vp0 + up1 * vp1 + up2 * vp2) * inv;
  };

  if (interior) {
    // ---- Phase B (interior): TDM tile loads, double-buffered ----
    // D# constant groups (per ISA 8.4-8.6): 4B elems, tile 18x10x3,
    // tensor strides 128 / 16384, normal (non-iterate, non-gather) mode.
    i32x8 g1;
    g1[0] = 0x00020000;                 // data_size = 2 (4 bytes)
    g1[1] = (int)(128u << 16);          // tensor_dim0[15:0]
    g1[2] = (int)(128u << 16);          // tensor_dim1[15:0]
    g1[3] = (int)(18u << 16);           // tile_dim0 = 18
    g1[4] = (int)((3u << 16) | 10u);    // tile_dim2 = 3 | tile_dim1 = 10
    g1[5] = 128;                        // tensor_dim0_stride = W
    g1[6] = (int)(16384u << 16);        // tensor_dim1_stride[15:0] = H*W
    g1[7] = 0;
    i32x4 g2; g2[0] = 8; g2[1] = 0; g2[2] = 0; g2[3] = 0;   // tensor_dim2 = D
    i32x4 g3; g3[0] = 0; g3[1] = 0; g3[2] = 0; g3[3] = 0;
    i32x8 g4;                           // extra arg of 6-arg builtin: zeros
    #pragma unroll
    for (int i = 0; i < 8; ++i) g4[i] = 0;

    auto issue_tdm = [&](int cc, int par, int cw) {
      const unsigned lds = (unsigned)(uintptr_t)&win[par][cw][0];
      const unsigned long long ga = (unsigned long long)(uintptr_t)
          (hb + (long)cc * DHW_ + (long)(d0 - 1) * HW_
              + (long)(y0 - 1) * W_ + (x0 - 1));
      u32x4 g0;
      g0[0] = 1u;                                       // count=1 user D#
      g0[1] = lds;                                      // lds_addr
      g0[2] = (unsigned)(ga & 0xFFFFFFFFull);           // global_addr lo
      g0[3] = (unsigned)((ga >> 32) & 0x1FFFFFFu) | 0x80000000u; // hi | type=2
      __builtin_amdgcn_tensor_load_to_lds(g0, g1, g2, g3, g4, 0);
    };

    if (t < 32) { issue_tdm(0, 0, 0); issue_tdm(1, 0, 1); }   // preload c0=0

    for (int c0 = 0; c0 < C_; c0 += 2) {
      const int cur = (c0 >> 1) & 1;
      if (t < 32) {
        if (c0 + 2 < C_) {
          issue_tdm(c0 + 2, cur ^ 1, 0);
          issue_tdm(c0 + 3, cur ^ 1, 1);
          __builtin_amdgcn_s_wait_tensorcnt(2);  // in-order: current pair done
        } else {
          __builtin_amdgcn_s_wait_tensorcnt(0);
        }
      }
      __syncthreads();                     // win[cur] visible to all waves
      dw_compute(win[cur][chalf], c0 + chalf);
      __syncthreads();                     // reads done before buffer reuse
    }
  } else {
    // ---- Phase B (boundary): clamped gather, async global->LDS ----
    for (int c0 = 0; c0 < C_; c0 += 2) {
      __syncthreads();   // protect window reuse
      for (int i = t; i < 2 * 540; i += 256) {
        const int cw   = i / 540;
        const int widx = i % 540;          // zz*180 + yy*18 + xx
        const int zz = widx / 180;
        const int rem = widx % 180;
        const int yy = rem / 18;
        const int xx = rem % 18;
        int dd = d0 + zz - 1; dd = dd < 0 ? 0 : (dd > D_ - 1 ? D_ - 1 : dd);
        int gy = y0 + yy - 1; gy = gy < 0 ? 0 : (gy > H_ - 1 ? H_ - 1 : gy);
        int gx = x0 + xx - 1; gx = gx < 0 ? 0 : (gx > W_ - 1 ? W_ - 1 : gx);
        const unsigned lds = (unsigned)(uintptr_t)&win[0][cw][widx];
        const unsigned long long ga = (unsigned long long)(uintptr_t)
            (hb + (long)(c0 + cw) * DHW_ + (long)dd * HW_ + gy * W_ + gx);
        asm volatile("global_load_async_to_lds_b32 %0, %1, off"
                     :: "v"(lds), "v"(ga) : "memory");
      }
      asm volatile("s_wait_asynccnt 0x0" ::: "memory");
      __syncthreads();
      if (c0 + 2 < C_)
        __builtin_prefetch(hb + (long)(c0 + 2 + chalf) * DHW_ + center, 0, 1);
      dw_compute(win[0][chalf], c0 + chalf);
    }
  }
  __syncthreads();

  // ---- Phase C: tail GEMM out[vox,o] = sum_c UV[vox,c]*tail_w[o,c] ----
  const int wave = t >> 5;
  const int lane = t & 31;
  const int half = lane >> 4;
  const int mrow = lane & 15;
  const int ntile = wave & 3;
  const int ocol = ntile * 16 + mrow;
  for (int mt = (wave >> 2); mt < 8; mt += 2) {
    v8f acc = {0.f, 0.f, 0.f, 0.f, 0.f, 0.f, 0.f, 0.f};
    #pragma unroll
    for (int k0 = 0; k0 < C_; k0 += 4) {
      const int ka = k0 + 2 * half;
      v2f a, bm;
      a.x  = uvt[(mt * 16 + mrow) * UVPITCH + ka];   // A[m,k] = UV[vox,c]
      a.y  = uvt[(mt * 16 + mrow) * UVPITCH + ka + 1];
      bm.x = tT[ka * WPITCH + ocol];                 // B[k,n] = tail_w[o,c]
      bm.y = tT[(ka + 1) * WPITCH + ocol];
      acc = __builtin_amdgcn_wmma_f32_16x16x4_f32(
          false, a, false, bm, (short)0, acc, false, false);
    }
    // D element (M,N): voxel = mt*16 + j + 8*half (=> ly=mt, lx=j+8*half)
    float* op = out + ((long)b * C_ + ocol) * DHW_ + (long)d0 * HW_
                    + (long)(y0 + mt) * W_ + x0 + 8 * half;
    #pragma unroll
    for (int j = 0; j < 8; ++j) op[j] = acc[j];
  }
}

extern "C" void kernel_launch(void* const* d_in, const int* in_sizes, int n_in,
                              void* d_out, int out_size, void* d_ws, size_t ws_size,
                              hipStream_t stream) {
  const float* x      = (const float*)d_in[0];
  const float* head_w = (const float*)d_in[1];
  const float* tail_w = (const float*)d_in[2];
  const float* U_w    = (const float*)d_in[3];
  const float* V_w    = (const float*)d_in[4];
  float* out = (float*)d_out;
  float* h   = (float*)d_ws;   // needs B*C*D*H*W*4 = 64 MiB scratch

  // Pass 1: one block per 128 contiguous voxels
  head_gemm_kernel<<<NVOX / 128, 256, 0, stream>>>(x, head_w, h);
  // Pass 2: one block per 8x16 spatial tile: 2*8*16*8 = 2048 blocks
  fused_lowrank_tail_kernel<<<B_ * D_ * (H_ / 8) * (W_ / 16), 256, 0, stream>>>(
      h, tail_w, U_w, V_w, out);
}